// Self_Attention_SMC_64553358459299
// MI455X (gfx1250) — compile-verified
//
#include <hip/hip_runtime.h>
#include <hip/hip_bf16.h>

// Shapes from the reference
#define Bx 8
#define Px 16
#define Sx 4096
#define Dx 512
#define Hx (Bx*Px)          // 128 heads (b,p flattened)

typedef __attribute__((ext_vector_type(2))) float v2f;
typedef __attribute__((ext_vector_type(4))) float v4f;
typedef __attribute__((ext_vector_type(8))) float v8f;

// ---------------------------------------------------------------------------
// Zero a float buffer (z accumulator)
// ---------------------------------------------------------------------------
__global__ void zero_f32(float* __restrict__ p, int n) {
  int i = blockIdx.x * blockDim.x + threadIdx.x;
  if (i < n) p[i] = 0.f;
}

// ---------------------------------------------------------------------------
// Projection GEMM via V_WMMA_F32_16X16X4_F32:  out = X(128x512) @ W(512x512) + b
// One wave (32 threads) per 16x16 output tile; 8 M-tiles x 32 N-tiles = 256 blocks.
// Optionally also scatters the result row-wise into a KV cache at `timestep`.
// A layout (16x4 f32): lane<16 -> K={k,k+1}, lane>=16 -> K={k+2,k+3} (ISA 7.12.2)
// B layout (4x16 f32): same half/K-pair indexing, N = lane&15
// C/D layout: VGPR r, lane l -> M = r + 8*(l>>4), N = l&15
// ---------------------------------------------------------------------------
__global__ __launch_bounds__(32) void proj_wmma(
    const float* __restrict__ X, const float* __restrict__ W,
    const float* __restrict__ bias, float* __restrict__ outM,
    float* __restrict__ cache, const int* __restrict__ tptr)
{
  const int lane = threadIdx.x;
  const int tile = blockIdx.x;
  const int m0   = (tile & 7) << 4;   // 8 M tiles
  const int n0   = (tile >> 3) << 4;  // 32 N tiles
  const int half = lane >> 4;         // 0: K pair {0,1}, 1: K pair {2,3}
  const int nr   = lane & 15;

  const float* __restrict__ arow = X + (size_t)(m0 + nr) * Dx;
  v8f c = {};
  for (int k = 0; k < Dx; k += 4) {
    const int ka = k + 2 * half;
    v2f a = *(const v2f*)(arow + ka);                 // A[m][ka], A[m][ka+1]
    v2f b;
    b.x = W[(size_t)ka * Dx + n0 + nr];               // B[ka][n]
    b.y = W[(size_t)(ka + 1) * Dx + n0 + nr];         // B[ka+1][n]
    c = __builtin_amdgcn_wmma_f32_16x16x4_f32(
        /*neg_a=*/false, a, /*neg_b=*/false, b,
        /*c_mod=*/(short)0, c, /*reuse_a=*/false, /*reuse_b=*/false);
  }

  const int t = *tptr;
  #pragma unroll
  for (int r = 0; r < 8; ++r) {
    const int m = m0 + r + (half << 3);
    const int n = n0 + nr;
    const float val = c[r] + bias[n];
    if (outM)  outM[(size_t)m * Dx + n] = val;
    if (cache) cache[(size_t)m * ((size_t)Sx * Dx) + (size_t)t * Dx + n] = val;
  }
}

// ---------------------------------------------------------------------------
// Stream K: copy cache rows to output (non-temporal, B128) and fuse the
// q·K dot product. One wave per row; lane handles 4 float4 chunks.
// Row s==timestep is already in Kout (fresh k): read it from there, skip copy.
// ---------------------------------------------------------------------------
__global__ __launch_bounds__(256) void k_stream_logits(
    const float* __restrict__ Kin, const float* __restrict__ q_ws,
    float* __restrict__ Kout, float* __restrict__ logits,
    const int* __restrict__ tptr)
{
  const int head  = blockIdx.y;
  const int chunk = blockIdx.x;
  const int t     = *tptr;
  const int wave  = threadIdx.x >> 5;
  const int lane  = threadIdx.x & 31;
  const size_t base = (size_t)head * ((size_t)Sx * Dx);

  // q fragment held in registers: lane covers d = 4*lane + 128*i, i=0..3
  v4f qv[4];
  const v4f* __restrict__ qh = (const v4f*)(q_ws + (size_t)head * Dx);
  #pragma unroll
  for (int i = 0; i < 4; ++i) qv[i] = qh[lane + 32 * i];

  const int s0 = chunk << 9;                       // 512 rows per block
  for (int s = s0 + wave; s < s0 + 512; s += 8) {  // 8 waves per block
    const size_t roff = base + (size_t)s * Dx;
    const bool fresh  = (s == t);
    const v4f* __restrict__ src = (const v4f*)((fresh ? Kout : Kin) + roff);
    v4f* __restrict__ dst = (v4f*)(Kout + roff);
    float acc = 0.f;
    #pragma unroll
    for (int i = 0; i < 4; ++i) {
      v4f x = __builtin_nontemporal_load(src + lane + 32 * i);
      if (!fresh) __builtin_nontemporal_store(x, dst + lane + 32 * i);
      acc += x.x * qv[i].x + x.y * qv[i].y + x.z * qv[i].z + x.w * qv[i].w;
    }
    #pragma unroll
    for (int off = 16; off > 0; off >>= 1)
      acc += __shfl_xor(acc, off, 32);
    if (lane == 0)
      logits[(size_t)head * Sx + s] = acc * 0.04419417382415922f; // 1/sqrt(512)
  }
}

// ---------------------------------------------------------------------------
// Masked softmax per head over S=4096; positions > timestep get weight 0.
// Writes the attn output directly.
// ---------------------------------------------------------------------------
__global__ __launch_bounds__(256) void softmax_head(
    const float* __restrict__ logits, float* __restrict__ attn,
    const int* __restrict__ tptr)
{
  __shared__ float red[256];
  const int head = blockIdx.x;
  const int t    = *tptr;
  const float* __restrict__ lh = logits + (size_t)head * Sx;

  float m = -3.4028235e38f;
  for (int s = threadIdx.x; s < Sx; s += 256)
    if (s <= t) m = fmaxf(m, lh[s]);
  red[threadIdx.x] = m;
  __syncthreads();
  for (int w = 128; w > 0; w >>= 1) {
    if (threadIdx.x < w) red[threadIdx.x] = fmaxf(red[threadIdx.x], red[threadIdx.x + w]);
    __syncthreads();
  }
  m = red[0];
  __syncthreads();

  float sum = 0.f;
  for (int s = threadIdx.x; s < Sx; s += 256)
    if (s <= t) sum += __expf(lh[s] - m);
  red[threadIdx.x] = sum;
  __syncthreads();
  for (int w = 128; w > 0; w >>= 1) {
    if (threadIdx.x < w) red[threadIdx.x] += red[threadIdx.x + w];
    __syncthreads();
  }
  const float inv = 1.f / red[0];

  for (int s = threadIdx.x; s < Sx; s += 256)
    attn[(size_t)head * Sx + s] = (s <= t) ? __expf(lh[s] - m) * inv : 0.f;
}

// ---------------------------------------------------------------------------
// Stream V: copy cache rows to output (non-temporal, B128) and fuse the
// attn-weighted accumulation into z (pre-projection), via f32 atomics.
// Thread covers a fixed float4 column f; two row-groups per block.
// ---------------------------------------------------------------------------
__global__ __launch_bounds__(256) void v_stream_accum(
    const float* __restrict__ Vin, const float* __restrict__ attn,
    float* __restrict__ Vout, float* __restrict__ zacc,
    const int* __restrict__ tptr)
{
  const int head  = blockIdx.y;
  const int chunk = blockIdx.x;
  const int t     = *tptr;
  const size_t base = (size_t)head * ((size_t)Sx * Dx);
  const int f    = threadIdx.x & 127;   // float4 column, d = 4f..4f+3
  const int rsel = threadIdx.x >> 7;    // 0/1: alternate rows
  const float* __restrict__ ah = attn + (size_t)head * Sx;

  v4f acc = {};
  const int s0 = chunk << 9;
  for (int s = s0 + rsel; s < s0 + 512; s += 2) {
    const float a = ah[s];
    const size_t roff = base + (size_t)s * Dx;
    const bool fresh  = (s == t);
    const v4f* __restrict__ src = (const v4f*)((fresh ? Vout : Vin) + roff);
    v4f x = __builtin_nontemporal_load(src + f);
    if (!fresh) __builtin_nontemporal_store(x, (v4f*)(Vout + roff) + f);
    acc += a * x;
  }
  float* zp = zacc + (size_t)head * Dx + 4 * f;
  atomicAdd(zp + 0, acc.x);
  atomicAdd(zp + 1, acc.y);
  atomicAdd(zp + 2, acc.z);
  atomicAdd(zp + 3, acc.w);
}

// ---------------------------------------------------------------------------
// Host-side launcher
// ---------------------------------------------------------------------------
extern "C" void kernel_launch(void* const* d_in, const int* in_sizes, int n_in,
                              void* d_out, int out_size, void* d_ws, size_t ws_size,
                              hipStream_t stream) {
  const float* x   = (const float*)d_in[0];   // (B,P,1,D) -> (128,512)
  const float* Kin = (const float*)d_in[1];   // (B,P,S,D)
  const float* Vin = (const float*)d_in[2];
  const float* Wq  = (const float*)d_in[3];
  const float* bq  = (const float*)d_in[4];
  const float* Wk  = (const float*)d_in[5];
  const float* bk  = (const float*)d_in[6];
  const float* Wv  = (const float*)d_in[7];
  const float* bv  = (const float*)d_in[8];
  const float* Wz  = (const float*)d_in[9];
  const float* bz  = (const float*)d_in[10];
  const int*  tptr = (const int*)d_in[11];

  // Output layout: z (128*512) | K (128*4096*512) | V (same) | attn (128*4096)
  float* out   = (float*)d_out;
  float* z_out = out;
  float* Kout  = out + (size_t)Hx * Dx;
  float* Vout  = Kout + (size_t)Hx * Sx * Dx;
  float* attn  = Vout + (size_t)Hx * Sx * Dx;

  // Workspace: q (128*512) | zacc (128*512) | logits (128*4096)
  float* ws     = (float*)d_ws;
  float* q_ws   = ws;
  float* zacc   = ws + (size_t)Hx * Dx;
  float* logits = ws + (size_t)2 * Hx * Dx;

  // 1) zero z accumulator (independent of everything else)
  zero_f32<<<dim3((Hx * Dx + 1023) / 1024), dim3(1024), 0, stream>>>(zacc, Hx * Dx);

  // 2) q/k/v projections (WMMA fp32). k/v go straight into the output caches
  //    at row `timestep`.
  proj_wmma<<<dim3(256), dim3(32), 0, stream>>>(x, Wq, bq, q_ws, nullptr, tptr);
  proj_wmma<<<dim3(256), dim3(32), 0, stream>>>(x, Wk, bk, nullptr, Kout, tptr);
  proj_wmma<<<dim3(256), dim3(32), 0, stream>>>(x, Wv, bv, nullptr, Vout, tptr);

  // 3) stream-copy K -> Kout fused with logits = q.K / sqrt(D)
  k_stream_logits<<<dim3(Sx / 512, Hx), dim3(256), 0, stream>>>(Kin, q_ws, Kout, logits, tptr);

  // 4) masked softmax -> attn output
  softmax_head<<<dim3(Hx), dim3(256), 0, stream>>>(logits, attn, tptr);

  // 5) stream-copy V -> Vout fused with z_pre = attn @ V
  v_stream_accum<<<dim3(Sx / 512, Hx), dim3(256), 0, stream>>>(Vin, attn, Vout, zacc, tptr);

  // 6) z = z_pre @ Wz + bz (WMMA fp32)
  proj_wmma<<<dim3(256), dim3(32), 0, stream>>>(zacc, Wz, bz, z_out, nullptr, tptr);

  (void)in_sizes; (void)n_in; (void)out_size; (void)ws_size;
}